// LeftRightMaxPooling_89593017794764
// MI455X (gfx1250) — compile-verified
//
#include <hip/hip_runtime.h>
#include <stdint.h>

// Row-wise cumulative max (cummax) over rows of length 2048, fp32.
// One wave32 per row. Async-tensor-era data path:
//   global --(global_load_async_to_lds_b128, coalesced)--> LDS (padded layout)
//   LDS --(ds_load_b128)--> regs --scan--> regs --(ds_store_b128)--> LDS
//   LDS --(global_store_async_from_lds_b128, coalesced)--> global
// Synchronized with the CDNA5 split counters (ASYNCcnt / DScnt).

#define ROW_LEN 2048
#define WAVES_PER_BLOCK 4
#define CHUNK 64                        // elements per lane (sequential scan)
#define CHUNK_DW (CHUNK + 4)            // 68 dwords: +16B pad kills bank conflicts
#define ROW_LDS_DW (32 * CHUNK_DW)      // 2176 dwords per row
// LDS per block: 4 * 2176 * 4 = 34,816 bytes

__global__ __launch_bounds__(WAVES_PER_BLOCK * 32)
void LeftRightMaxPooling_89593017794764_kernel(const float* __restrict__ src,
                                               float* __restrict__ dst,
                                               int rows) {
  __shared__ float lds[WAVES_PER_BLOCK * ROW_LDS_DW];

  const int tid  = threadIdx.x;
  const int wave = tid >> 5;
  const int lane = tid & 31;
  const int row  = blockIdx.x * WAVES_PER_BLOCK + wave;
  if (row >= rows) return;  // whole-wave uniform

  const uint64_t row_src = (uint64_t)(uintptr_t)(src + (size_t)row * ROW_LEN);
  const uint64_t row_dst = (uint64_t)(uintptr_t)(dst + (size_t)row * ROW_LEN);

  float* wrow = &lds[wave * ROW_LDS_DW];
  // Low 32 bits of a flat LDS pointer are the workgroup-relative LDS offset
  // (aperture lives in the high 32 bits) -- exactly what the async ops want.
  const unsigned wave_lds_base = (unsigned)(uintptr_t)wrow;

  // ---- Stage row into LDS: coalesced global reads, padded LDS scatter ----
  // float4 group g covers row bytes [g*16, g*16+16); chunk = g>>4 (64 elems),
  // LDS byte offset = chunk*272 + (g&15)*16.
#pragma unroll
  for (int j = 0; j < 16; ++j) {
    const unsigned g      = (unsigned)(j * 32 + lane);
    const unsigned ldsoff = wave_lds_base + (g >> 4) * (CHUNK_DW * 4) + (g & 15u) * 16u;
    const uint64_t gaddr  = row_src + (uint64_t)g * 16u;
    asm volatile("global_load_async_to_lds_b128 %0, %1, off"
                 :: "v"(ldsoff), "v"(gaddr)
                 : "memory");
  }
  asm volatile("s_wait_asynccnt 0x0" ::: "memory");

  // ---- Pass 1: per-lane sequential inclusive max-scan of its 64 elems ----
  const float4* lp = reinterpret_cast<const float4*>(wrow + lane * CHUNK_DW);
  float4 vals[16];
  float run = -__builtin_inff();
#pragma unroll
  for (int t = 0; t < 16; ++t) {
    float4 v = lp[t];
    run = fmaxf(run, v.x); v.x = run;
    run = fmaxf(run, v.y); v.y = run;
    run = fmaxf(run, v.z); v.z = run;
    run = fmaxf(run, v.w); v.w = run;
    vals[t] = v;
  }

  // ---- Wave-wide exclusive max-scan of per-lane totals (wave32) ----
  float inc = run;
#pragma unroll
  for (int d = 1; d < 32; d <<= 1) {
    float o = __shfl_up(inc, (unsigned)d, 32);
    if (lane >= d) inc = fmaxf(inc, o);
  }
  float prefix = __shfl_up(inc, 1u, 32);
  if (lane == 0) prefix = -__builtin_inff();

  // ---- Pass 2: fold in lane prefix, write results back to LDS ----
  float4* sp = reinterpret_cast<float4*>(wrow + lane * CHUNK_DW);
#pragma unroll
  for (int t = 0; t < 16; ++t) {
    float4 v = vals[t];
    v.x = fmaxf(v.x, prefix);
    v.y = fmaxf(v.y, prefix);
    v.z = fmaxf(v.z, prefix);
    v.w = fmaxf(v.w, prefix);
    sp[t] = v;
  }
  // Make the ds_stores visible to the async-store engine.
  asm volatile("s_wait_dscnt 0x0" ::: "memory");

  // ---- Drain LDS -> global: coalesced global writes ----
#pragma unroll
  for (int j = 0; j < 16; ++j) {
    const unsigned g      = (unsigned)(j * 32 + lane);
    const unsigned ldsoff = wave_lds_base + (g >> 4) * (CHUNK_DW * 4) + (g & 15u) * 16u;
    const uint64_t gaddr  = row_dst + (uint64_t)g * 16u;
    asm volatile("global_store_async_from_lds_b128 %0, %1, off"
                 :: "v"(gaddr), "v"(ldsoff)
                 : "memory");
  }
  asm volatile("s_wait_asynccnt 0x0" ::: "memory");
}

extern "C" void kernel_launch(void* const* d_in, const int* in_sizes, int n_in,
                              void* d_out, int out_size, void* d_ws, size_t ws_size,
                              hipStream_t stream) {
  (void)n_in; (void)d_ws; (void)ws_size; (void)out_size;
  const float* x = (const float*)d_in[0];
  float* out     = (float*)d_out;
  const int rows = in_sizes[0] / ROW_LEN;          // 16384 for (8,1,2048,2048)
  const int blocks = (rows + WAVES_PER_BLOCK - 1) / WAVES_PER_BLOCK;
  LeftRightMaxPooling_89593017794764_kernel<<<blocks, WAVES_PER_BLOCK * 32, 0, stream>>>(
      x, out, rows);
}